// BaseModel_31104153158101
// MI455X (gfx1250) — compile-verified
//
#include <hip/hip_runtime.h>

#ifndef __has_builtin
#define __has_builtin(x) 0
#endif

#define BLOCK 64          // threads per block = rays per block (2 waves on wave32)
#define CH 1920           // samples (float4 rows) staged per chunk: 30 KB
#define CHUNK_OPS (CH / BLOCK)  // async b128 instructions per wave per chunk = 30

#if __has_builtin(__builtin_amdgcn_global_load_async_to_lds_b128)
#define HAVE_ASYNC_LDS 1
#else
#define HAVE_ASYNC_LDS 0
#endif

typedef int v4i __attribute__((vector_size(16)));
#define AS1 __attribute__((address_space(1)))
#define AS3 __attribute__((address_space(3)))

template <int N>
__device__ __forceinline__ void wait_asynccnt_le() {
#if __has_builtin(__builtin_amdgcn_s_wait_asynccnt)
  __builtin_amdgcn_s_wait_asynccnt(N);
#else
  asm volatile("s_wait_asynccnt %0" : : "i"(N) : "memory");
#endif
}

__device__ __forceinline__ int lower_bound_i32(const int* __restrict__ a, int n, int key) {
  int lo = 0, hi = n;
  while (lo < hi) {
    int mid = (lo + hi) >> 1;
    if (a[mid] < key) lo = mid + 1; else hi = mid;
  }
  return lo;
}

__device__ __forceinline__ void issue_chunk(const float4* __restrict__ src, int c0,
                                            float4* sbuf, int tid) {
#if HAVE_ASYNC_LDS
  float4* gsrc = const_cast<float4*>(src);
#pragma unroll
  for (int q = tid; q < CH; q += BLOCK) {
    // async DMA: global (AS1) -> LDS (AS3), 16B per lane, coalesced across lanes
    __builtin_amdgcn_global_load_async_to_lds_b128(
        (AS1 v4i*)(gsrc + c0 + q),
        (AS3 v4i*)(sbuf + q),
        0, 0);
  }
#else
  for (int q = tid; q < CH; q += BLOCK) sbuf[q] = src[c0 + q];
#endif
}

__global__ void __launch_bounds__(BLOCK)
segsum_rays_kernel(const float4* __restrict__ src, const int* __restrict__ idx,
                   float4* __restrict__ out, int n_samples, int n_rays) {
  __shared__ float4 sbuf[2][CH];      // 61440 B double buffer
  __shared__ int sLo[BLOCK + 1];      // segment boundaries for this block's rays

  const int tid  = threadIdx.x;
  const int ray0 = blockIdx.x * BLOCK;
  const int r    = ray0 + tid;

  // lower bound of my ray; thread BLOCK-1 also finds the block's upper bound
  int myLo = (r < n_rays) ? lower_bound_i32(idx, n_samples, r) : n_samples;
  sLo[tid] = myLo;
  if (tid == BLOCK - 1)
    sLo[BLOCK] = lower_bound_i32(idx, n_samples, ray0 + BLOCK);
  __syncthreads();

  const int myHi    = (r < n_rays) ? sLo[tid + 1] : n_samples;  // hi[r] == lo[r+1]
  const int blockLo = sLo[0];
  const int blockHi = sLo[BLOCK];

  const int cBase = blockLo;
  const int full  = (blockHi - cBase) / CH;   // block-uniform

  float4 acc0 = make_float4(0.f, 0.f, 0.f, 0.f);
  float4 acc1 = make_float4(0.f, 0.f, 0.f, 0.f);

  if (full > 0) issue_chunk(src, cBase, &sbuf[0][0], tid);

  for (int k = 0; k < full; ++k) {
    if (k + 1 < full) issue_chunk(src, cBase + (k + 1) * CH, &sbuf[(k + 1) & 1][0], tid);
#if HAVE_ASYNC_LDS
    if (k + 1 < full) wait_asynccnt_le<CHUNK_OPS>();   // chunk k retired, k+1 in flight
    else              wait_asynccnt_le<0>();
#endif
    __syncthreads();   // LDS chunk k visible to all waves

    const float4* buf = &sbuf[k & 1][0];
    const int c0 = cBase + k * CH;
    int a = (myLo > c0) ? myLo : c0;
    int b = (myHi < c0 + CH) ? myHi : (c0 + CH);
    int i = a;
    for (; i + 1 < b; i += 2) {
      float4 v0 = buf[i - c0];
      float4 v1 = buf[i + 1 - c0];
      acc0.x += v0.x; acc0.y += v0.y; acc0.z += v0.z; acc0.w += v0.w;
      acc1.x += v1.x; acc1.y += v1.y; acc1.z += v1.z; acc1.w += v1.w;
    }
    if (i < b) {
      float4 v = buf[i - c0];
      acc0.x += v.x; acc0.y += v.y; acc0.z += v.z; acc0.w += v.w;
    }
    __syncthreads();   // everyone done reading before this buffer is refilled
  }

  // tail of the block's range (< CH samples): read directly from global (L2-hot)
  {
    int t0 = cBase + full * CH;
    if (myLo > t0) t0 = myLo;
    for (int i = t0; i < myHi; ++i) {
      float4 v = src[i];
      acc0.x += v.x; acc0.y += v.y; acc0.z += v.z; acc0.w += v.w;
    }
  }

  if (r < n_rays) {
    float4 res = make_float4(acc0.x + acc1.x, acc0.y + acc1.y,
                             acc0.z + acc1.z, acc0.w + acc1.w);
    out[r] = res;   // exactly one writer per ray: deterministic, no memset needed
  }
}

extern "C" void kernel_launch(void* const* d_in, const int* in_sizes, int n_in,
                              void* d_out, int out_size, void* d_ws, size_t ws_size,
                              hipStream_t stream) {
  (void)n_in; (void)d_ws; (void)ws_size;
  const float4* src = (const float4*)d_in[0];       // (N, 4) float32 rows
  const int*    idx = (const int*)d_in[1];          // sorted ray ids (int32 per harness)
  float4*       out = (float4*)d_out;               // (n_rays, 4) float32

  const int n_samples = in_sizes[1];
  const int n_rays    = out_size / 4;

  const int blocks = (n_rays + BLOCK - 1) / BLOCK;
  segsum_rays_kernel<<<blocks, BLOCK, 0, stream>>>(src, idx, out, n_samples, n_rays);
}